// MultiHeadAttention_16647293239562
// MI455X (gfx1250) — compile-verified
//
#include <hip/hip_runtime.h>
#include <math.h>
#include <stdint.h>

typedef float v2f __attribute__((ext_vector_type(2)));
typedef float v4f __attribute__((ext_vector_type(4)));
typedef float v8f __attribute__((ext_vector_type(8)));
typedef int   v4i __attribute__((ext_vector_type(4)));

#define WMMA_F32(a, b, c) \
  __builtin_amdgcn_wmma_f32_16x16x4_f32(false, (a), false, (b), (short)0, (c), false, false)

// gfx1250 async global->LDS (ASYNCcnt-tracked, no VGPR round trip).
#if defined(__HIP_DEVICE_COMPILE__) &&                                \
    __has_builtin(__builtin_amdgcn_global_load_async_to_lds_b128) && \
    __has_builtin(__builtin_amdgcn_s_wait_asynccnt)
#define USE_ASYNC_LDS 1
#else
#define USE_ASYNC_LDS 0
#endif
// per-lane b32 scatter variant (used for transposed tile fills)
#if USE_ASYNC_LDS && __has_builtin(__builtin_amdgcn_global_load_async_to_lds_b32)
#define USE_ASYNC_B32 1
#else
#define USE_ASYNC_B32 0
#endif

#if USE_ASYNC_LDS
typedef v4i __attribute__((address_space(1)))* as1_v4i;  // global int4*
typedef v4i __attribute__((address_space(3)))* as3_v4i;  // LDS int4*
#endif
#if USE_ASYNC_B32
typedef int __attribute__((address_space(1)))* as1_i32;  // global int*
typedef int __attribute__((address_space(3)))* as3_i32;  // LDS int*
#endif

__device__ __forceinline__ void async_ld16(const float* g, float* l) {
#if USE_ASYNC_LDS
  __builtin_amdgcn_global_load_async_to_lds_b128(
      (as1_v4i)(uintptr_t)g, (as3_v4i)l, 0, 0);
#endif
}
__device__ __forceinline__ void async_ld4(const float* g, float* l) {
#if USE_ASYNC_B32
  __builtin_amdgcn_global_load_async_to_lds_b32(
      (as1_i32)(uintptr_t)g, (as3_i32)l, 0, 0);
#endif
}
__device__ __forceinline__ void async_wait0() {
#if USE_ASYNC_LDS
  __builtin_amdgcn_s_wait_asynccnt(0);
#endif
}

static constexpr int Bsz = 2, Lseq = 2048, Dmod = 1024, NH = 16, DHd = 64;

// ---------------------------------------------------------------------------
// fp32 GEMM: C[M,N] = A[M,K] * B[K,N], row-major, dims % 64 == 0, K % 32 == 0.
// 128 threads (4 waves), 64x64 tile, K-step 32, double-buffered LDS.
// BOTH tiles stored with k innermost ([m][k] / [n][k], ld 36): fragment pairs
// are adjacent -> pure ds_load_b64, no VGPR repacking movs competing with
// v_wmma issue. B tile is filled transposed via per-lane async b32 scatter
// (global side stays fully coalesced; LDS scatter handled by the async unit).
// ---------------------------------------------------------------------------
__global__ __launch_bounds__(128) void gemm_f32(const float* __restrict__ A,
                                                const float* __restrict__ B,
                                                float* __restrict__ C,
                                                int M, int N, int K) {
  __shared__ __attribute__((aligned(16))) float As[2][64][36];  // [m][k]
  __shared__ __attribute__((aligned(16))) float Bt[2][64][36];  // [n][k]
  const int tid = threadIdx.x;
  const int w = tid >> 5, lane = tid & 31, r = lane & 15, h = lane >> 4;
  const int m0 = blockIdx.y * 64, n0 = blockIdx.x * 64;
  const int atx = (tid & 7) * 4, aty = tid >> 3;  // A: 8x16 thr, v4 cols
  const int bn = tid & 63, bk = tid >> 6;         // B: 64 n-lanes x 2 k-rows
  const int bcol = (tid & 15) * 4, bty = tid >> 4;  // B fallback mapping
  const size_t aRS = (size_t)16 * K;  // 16-row stride in A
  const size_t bRS = (size_t)8 * N;   // 8-row stride in B
  const float* aP = A + (size_t)(m0 + aty) * K + atx;
  const float* bP = B + (size_t)bty * N + n0 + bcol;
  const float* bP2 = B + (size_t)bk * N + n0 + bn;

  v8f acc[4] = {};
#if !(USE_ASYNC_LDS && USE_ASYNC_B32)
  v4f ra[4], rb[4];
#endif

  auto issue = [&](int buf, int k0) {
#if USE_ASYNC_LDS && USE_ASYNC_B32
#pragma unroll
    for (int e = 0; e < 4; ++e)
      async_ld16(aP + k0 + e * aRS, &As[buf][aty + 16 * e][atx]);
#pragma unroll
    for (int e = 0; e < 16; ++e)  // transposed scatter fill of B tile
      async_ld4(bP2 + (size_t)(k0 + 2 * e) * N, &Bt[buf][bn][bk + 2 * e]);
#else
#pragma unroll
    for (int e = 0; e < 4; ++e) ra[e] = *(const v4f*)(aP + k0 + e * aRS);
#pragma unroll
    for (int e = 0; e < 4; ++e)
      rb[e] = *(const v4f*)(bP + (size_t)k0 * N + e * bRS);
    (void)buf;
#endif
  };
  auto commit = [&](int buf) {
#if USE_ASYNC_LDS && USE_ASYNC_B32
    async_wait0();
    (void)buf;
#else
#pragma unroll
    for (int e = 0; e < 4; ++e) *(v4f*)&As[buf][aty + 16 * e][atx] = ra[e];
#pragma unroll
    for (int e = 0; e < 4; ++e)
#pragma unroll
      for (int c = 0; c < 4; ++c)
        Bt[buf][bcol + c][bty + 8 * e] = rb[e][c];  // transpose on store
#endif
  };

  const int nsteps = K >> 5;
  issue(0, 0);
  commit(0);
  __syncthreads();

  for (int s = 0; s < nsteps; ++s) {
    const int cur = s & 1;
    if (s + 1 < nsteps) issue(cur ^ 1, (s + 1) << 5);  // overlap next tile
#pragma unroll
    for (int kk = 0; kk < 32; kk += 4) {
      v2f a;
      a.x = As[cur][16 * w + r][kk + 2 * h];
      a.y = As[cur][16 * w + r][kk + 2 * h + 1];
#pragma unroll
      for (int t = 0; t < 4; ++t) {
        v2f b;  // B[k][n] = Bt[n][k]: adjacent pair -> single b64
        b.x = Bt[cur][16 * t + r][kk + 2 * h];
        b.y = Bt[cur][16 * t + r][kk + 2 * h + 1];
        acc[t] = WMMA_F32(a, b, acc[t]);
      }
    }
    if (s + 1 < nsteps) commit(cur ^ 1);
    __syncthreads();
  }
  (void)M;

#pragma unroll
  for (int t = 0; t < 4; ++t)
#pragma unroll
    for (int i = 0; i < 8; ++i)
      C[(size_t)(m0 + 16 * w + i + 8 * h) * N + n0 + 16 * t + r] = acc[t][i];
}

// ---------------------------------------------------------------------------
// RoPE + permute: qkv_raw [B,L,3,H,DH] -> Q,K (roped), V as [B,H,L,DH].
// Softmax scale 1/sqrt(DH)=0.125 folded into Q. libm sin/cos kept (angles up
// to 2047 rad need full range reduction).
// ---------------------------------------------------------------------------
__global__ void rope_permute(const float* __restrict__ qkv,
                             float* __restrict__ Q, float* __restrict__ K,
                             float* __restrict__ V) {
  int idx = blockIdx.x * 256 + threadIdx.x;  // over B*L*H*DH
  int dh = idx & 63;
  int tmp = idx >> 6;
  int hh = tmp & 15;
  tmp >>= 4;
  int l = tmp & 2047;
  int b = tmp >> 11;

  size_t base = ((size_t)(b * Lseq + l) * 3) * Dmod + (size_t)hh * DHd;
  float q = qkv[base + dh];
  float k = qkv[base + Dmod + dh];
  float v = qkv[base + 2 * Dmod + dh];
  float qp = (dh < 32) ? -qkv[base + dh + 32] : qkv[base + dh - 32];
  float kp = (dh < 32) ? -qkv[base + Dmod + dh + 32] : qkv[base + Dmod + dh - 32];

  float inv = powf(10000.0f, -(float)(2 * (dh & 31)) * (1.0f / 64.0f));
  float ang = (float)l * inv;
  float c = cosf(ang), s = sinf(ang);

  size_t o = ((size_t)(b * NH + hh) * Lseq + l) * DHd + dh;
  Q[o] = (q * c + qp * s) * 0.125f;
  K[o] = k * c + kp * s;
  V[o] = v;
}

// ---------------------------------------------------------------------------
// Flash attention, fp32 WMMA. One block = 64 query rows of one (b,h).
// Q/K tiles natural [row][d] (ld 68 -> bank 4r strided reads; K^T adjacency
// gives b64 B-frag loads). V is staged TRANSPOSED [d][j] into the shared KV
// buffer so PV B-frags are adjacent too (identical indexing to the S phase).
// ---------------------------------------------------------------------------
__global__ __launch_bounds__(128) void flash_attn(const float* __restrict__ Q,
                                                  const float* __restrict__ K,
                                                  const float* __restrict__ V,
                                                  float* __restrict__ O) {
  __shared__ __attribute__((aligned(16))) float Qs[64][68];
  __shared__ __attribute__((aligned(16))) float KVs[64][68];
  __shared__ __attribute__((aligned(16))) float Ps[4][16][68];  // P as [m][j]

  const int tid = threadIdx.x;
  const int w = tid >> 5, lane = tid & 31, r = lane & 15, h = lane >> 4;
  const int bh = blockIdx.y, q0 = blockIdx.x * 64;
  const float* Qb = Q + (size_t)bh * Lseq * DHd;
  const float* Kb = K + (size_t)bh * Lseq * DHd;
  const float* Vb = V + (size_t)bh * Lseq * DHd;
  const int tx = (tid & 15) * 4;  // v4 along d (64)
  const int ty = tid >> 4;        // 8 rows per pass
  const int vd = tid & 63, vj = tid >> 6;  // V transposed-scatter mapping

  // Q tile, natural [row][d]
#if USE_ASYNC_LDS
#pragma unroll
  for (int e = 0; e < 8; ++e)
    async_ld16(Qb + (size_t)(q0 + ty + 8 * e) * DHd + tx, &Qs[ty + 8 * e][tx]);
#else
  {
    v4f rq[8];
#pragma unroll
    for (int e = 0; e < 8; ++e)
      rq[e] = *(const v4f*)(Qb + (size_t)(q0 + ty + 8 * e) * DHd + tx);
#pragma unroll
    for (int e = 0; e < 8; ++e) *(v4f*)&Qs[ty + 8 * e][tx] = rq[e];
  }
#endif

  v8f o_acc[4] = {};
  float m_i[8], l_i[8];
#pragma unroll
  for (int i = 0; i < 8; ++i) { m_i[i] = -3.0e38f; l_i[i] = 0.0f; }

  for (int kc = 0; kc < Lseq; kc += 64) {
    __syncthreads();  // prior chunk's V readers done (also orders Q vs. load)
    // ---- K chunk into KVs, natural [j][d] ----
#if USE_ASYNC_LDS
#pragma unroll
    for (int e = 0; e < 8; ++e)
      async_ld16(Kb + (size_t)(kc + ty + 8 * e) * DHd + tx,
                 &KVs[ty + 8 * e][tx]);
    async_wait0();
#else
    {
      v4f rk[8];
#pragma unroll
      for (int e = 0; e < 8; ++e)
        rk[e] = *(const v4f*)(Kb + (size_t)(kc + ty + 8 * e) * DHd + tx);
#pragma unroll
      for (int e = 0; e < 8; ++e) *(v4f*)&KVs[ty + 8 * e][tx] = rk[e];
    }
#endif
    __syncthreads();

    // ---- S = (Q*scale) K^T, K-dim = DH = 64 ----
    v8f s_acc[4] = {};
#pragma unroll
    for (int kk = 0; kk < 64; kk += 4) {
      v2f a;
      a.x = Qs[16 * w + r][kk + 2 * h];
      a.y = Qs[16 * w + r][kk + 2 * h + 1];
#pragma unroll
      for (int t = 0; t < 4; ++t) {
        v2f b;  // B[k][n] = K[n][k] -> adjacent in KVs rows
        b.x = KVs[16 * t + r][kk + 2 * h];
        b.y = KVs[16 * t + r][kk + 2 * h + 1];
        s_acc[t] = WMMA_F32(a, b, s_acc[t]);
      }
    }

    // ---- online softmax (row i+8h lives on the 16 lanes sharing h) ----
#pragma unroll
    for (int i = 0; i < 8; ++i) {
      float mx = fmaxf(fmaxf(s_acc[0][i], s_acc[1][i]),
                       fmaxf(s_acc[2][i], s_acc[3][i]));
      mx = fmaxf(mx, __shfl_xor(mx, 1, 32));
      mx = fmaxf(mx, __shfl_xor(mx, 2, 32));
      mx = fmaxf(mx, __shfl_xor(mx, 4, 32));
      mx = fmaxf(mx, __shfl_xor(mx, 8, 32));
      float m_new = fmaxf(m_i[i], mx);
      float alpha = __expf(m_i[i] - m_new);
      float rs = 0.0f;
#pragma unroll
      for (int t = 0; t < 4; ++t) {
        float p = __expf(s_acc[t][i] - m_new);
        s_acc[t][i] = p;
        rs += p;
      }
      rs += __shfl_xor(rs, 1, 32);
      rs += __shfl_xor(rs, 2, 32);
      rs += __shfl_xor(rs, 4, 32);
      rs += __shfl_xor(rs, 8, 32);
      l_i[i] = l_i[i] * alpha + rs;
      m_i[i] = m_new;
#pragma unroll
      for (int t = 0; t < 4; ++t) o_acc[t][i] *= alpha;
    }

    // ---- stage P (C-layout -> A-layout [m][j]) in per-wave LDS ----
#pragma unroll
    for (int t = 0; t < 4; ++t)
#pragma unroll
      for (int i = 0; i < 8; ++i)
        Ps[w][i + 8 * h][16 * t + r] = s_acc[t][i];

    __syncthreads();  // all waves done reading K from KVs
    // ---- V chunk into KVs, TRANSPOSED [d][j] ----
#if USE_ASYNC_LDS && USE_ASYNC_B32
#pragma unroll
    for (int e = 0; e < 32; ++e)
      async_ld4(Vb + (size_t)(kc + vj + 2 * e) * DHd + vd,
                &KVs[vd][vj + 2 * e]);
    async_wait0();
#else
    {
      v4f rv[8];
#pragma unroll
      for (int e = 0; e < 8; ++e)
        rv[e] = *(const v4f*)(Vb + (size_t)(kc + ty + 8 * e) * DHd + tx);
#pragma unroll
      for (int e = 0; e < 8; ++e)
#pragma unroll
        for (int c = 0; c < 4; ++c)
          KVs[tx + c][ty + 8 * e] = rv[e][c];  // transpose on store
    }
#endif
    __syncthreads();

    // ---- O += P V, K-dim = 64 (B[k][n] = Vt[n][k]: adjacent pairs) ----
#pragma unroll
    for (int kk = 0; kk < 64; kk += 4) {
      v2f a;
      a.x = Ps[w][r][kk + 2 * h];
      a.y = Ps[w][r][kk + 2 * h + 1];
#pragma unroll
      for (int t = 0; t < 4; ++t) {
        v2f b;
        b.x = KVs[16 * t + r][kk + 2 * h];
        b.y = KVs[16 * t + r][kk + 2 * h + 1];
        o_acc[t] = WMMA_F32(a, b, o_acc[t]);
      }
    }
  }

  // normalize + write attn_out as [B, L, H, DH] (== [B*L, D] for Wo GEMM)
  const int b = bh >> 4, hh = bh & 15;
#pragma unroll
  for (int i = 0; i < 8; ++i) {
    float il = 1.0f / l_i[i];
#pragma unroll
    for (int t = 0; t < 4; ++t) {
      int lq = q0 + 16 * w + i + 8 * h;
      O[((size_t)(b * Lseq + lq) * NH + hh) * DHd + 16 * t + r] =
          o_acc[t][i] * il;
    }
  }
}

// ---------------------------------------------------------------------------
extern "C" void kernel_launch(void* const* d_in, const int* in_sizes, int n_in,
                              void* d_out, int out_size, void* d_ws,
                              size_t ws_size, hipStream_t stream) {
  (void)in_sizes; (void)n_in; (void)out_size; (void)ws_size;
  const float* x    = (const float*)d_in[0];
  const float* Wqkv = (const float*)d_in[1];
  const float* Wo   = (const float*)d_in[2];
  float* out = (float*)d_out;
  float* ws  = (float*)d_ws;

  const size_t perT = (size_t)Bsz * NH * Lseq * DHd;  // 4M floats
  float* Qr       = ws;
  float* Kr       = ws + perT;
  float* Vt       = ws + 2 * perT;
  float* qkv_raw  = ws + 3 * perT;   // [B*L, 3D]
  float* attn_out = qkv_raw;         // reuse after rope consumed qkv_raw

  gemm_f32<<<dim3(3 * Dmod / 64, Bsz * Lseq / 64), 128, 0, stream>>>(
      x, Wqkv, qkv_raw, Bsz * Lseq, 3 * Dmod, Dmod);

  rope_permute<<<dim3((Bsz * Lseq * NH * DHd) / 256), 256, 0, stream>>>(
      qkv_raw, Qr, Kr, Vt);

  flash_attn<<<dim3(Lseq / 64, Bsz * NH), 128, 0, stream>>>(Qr, Kr, Vt,
                                                            attn_out);

  gemm_f32<<<dim3(Dmod / 64, Bsz * Lseq / 64), 128, 0, stream>>>(
      attn_out, Wo, out, Bsz * Lseq, Dmod, Dmod);
}